// DiceLoss_19550691131598
// MI455X (gfx1250) — compile-verified
//
#include <hip/hip_runtime.h>
#include <hip/hip_bf16.h>

typedef float v2f __attribute__((ext_vector_type(2)));
typedef float v8f __attribute__((ext_vector_type(8)));

#define BATCH 64
#define KSLOT 16
#define NPIX  65536
#define EPSC  1e-8f
#define SMOOTH 0.1f

// Kernel 1: per (batch, chunk) block computes partial Gram (16x16) via
// v_wmma_f32_16x16x4_f32 and partial row sums over its N-slice.
__global__ __launch_bounds__(256) void dice_gram_kernel(
    const float* __restrict__ am,
    float* __restrict__ Gp,      // [BATCH][nchunks][256]
    float* __restrict__ Sp,      // [BATCH][nchunks][16]
    int nchunks)
{
    const int b     = blockIdx.y;
    const int chunk = blockIdx.x;
    const int tid   = threadIdx.x;
    const int wave  = tid >> 5;
    const int lane  = tid & 31;
    const int row   = lane & 15;     // matrix row this lane holds (M and N role)
    const int hi    = lane >> 4;     // 0 -> K=0,1 ; 1 -> K=2,3

    const int colsPerBlock = NPIX / nchunks;
    const int colsPerWave  = colsPerBlock >> 3;   // 8 waves per block
    const int n0 = chunk * colsPerBlock + wave * colsPerWave;

    // lane L reads row (L&15), cols n0 + 2*(L>>4) + {0,1}, stepping by 4.
    const float* p = am + ((size_t)b * KSLOT + row) * (size_t)NPIX + n0 + 2 * hi;

    v8f   acc  = {};      // 16x16 f32 accumulator (C/D), 8 VGPRs
    float rsum = 0.0f;    // partial row sum for row (lane&15)

    #pragma unroll 8
    for (int n = 0; n < colsPerWave; n += 4) {
        v2f a = *(const v2f*)(p + n);          // global_load_b64, A-layout for 16x4 f32
        rsum += a.x + a.y;
        // Gram: D = A * A^T + C ; f32 A/B layouts coincide for the same data.
        acc = __builtin_amdgcn_wmma_f32_16x16x4_f32(
            /*neg_a=*/false, a, /*neg_b=*/false, a,
            /*c_mod=*/(short)0, acc, /*reuse_a=*/false, /*reuse_b=*/false);
    }

    __shared__ float ldsC[8][32][8];   // 8 KB
    __shared__ float ldsS[8][32];      // 1 KB
    #pragma unroll
    for (int j = 0; j < 8; ++j) ldsC[wave][lane][j] = acc[j];
    ldsS[wave][lane] = rsum;
    __syncthreads();

    // Reduce the 8 wave-accumulators: thread t owns C slot (vgpr j = t>>5, lane L = t&31).
    {
        const int j = tid >> 5;
        const int L = tid & 31;
        float g = 0.0f;
        #pragma unroll
        for (int w = 0; w < 8; ++w) g += ldsC[w][L][j];
        // C layout: element (m, n) lives at vgpr j = m%8, lane = (m/8)*16 + n
        const int m  = j + 8 * (L >> 4);
        const int nn = L & 15;
        Gp[((size_t)b * nchunks + chunk) * 256 + m * 16 + nn] = g;
    }
    // Row sums: rows split across lane pairs (L, L+16) and 8 waves.
    if (tid < 16) {
        float s = 0.0f;
        #pragma unroll
        for (int w = 0; w < 8; ++w) s += ldsS[w][tid] + ldsS[w][tid + 16];
        Sp[((size_t)b * nchunks + chunk) * 16 + tid] = s;
    }
}

// Kernel 2: one block; reduce chunks, apply dice formula, batch-mean,
// upper-triangle sum, divide by K*(K-1)/2 = 120.
__global__ __launch_bounds__(256) void dice_finalize_kernel(
    const float* __restrict__ Gp,
    const float* __restrict__ Sp,
    float* __restrict__ out,
    int nchunks)
{
    __shared__ float S[BATCH * KSLOT];   // 4 KB
    __shared__ float red[256];
    const int tid = threadIdx.x;

    for (int i = tid; i < BATCH * KSLOT; i += 256) {
        const int b = i >> 4, k = i & 15;
        float s = 0.0f;
        for (int c = 0; c < nchunks; ++c) s += Sp[((size_t)b * nchunks + c) * 16 + k];
        S[i] = s;
    }
    __syncthreads();

    const int k = tid >> 4, l = tid & 15;
    float acc = 0.0f;
    for (int b = 0; b < BATCH; ++b) {
        float g = 0.0f;
        for (int c = 0; c < nchunks; ++c) g += Gp[((size_t)b * nchunks + c) * 256 + tid];
        const float sk = S[b * 16 + k];
        const float sl = S[b * 16 + l];
        const float dk = sk + EPSC;
        const float dl = sl + EPSC;
        const float gram = g / (dk * dl);          // normalized Gram
        const float sumk = sk / dk;
        const float suml = sl / dl;
        acc += (2.0f * gram + SMOOTH) / (sumk + suml + SMOOTH);
    }
    red[tid] = (k < l) ? (acc / (float)BATCH) : 0.0f;
    __syncthreads();
    for (int s2 = 128; s2 > 0; s2 >>= 1) {
        if (tid < s2) red[tid] += red[tid + s2];
        __syncthreads();
    }
    if (tid == 0) out[0] = red[0] / 120.0f;
}

extern "C" void kernel_launch(void* const* d_in, const int* in_sizes, int n_in,
                              void* d_out, int out_size, void* d_ws, size_t ws_size,
                              hipStream_t stream) {
    const float* am  = (const float*)d_in[0];
    float*       out = (float*)d_out;

    // Pick N-split per batch by available workspace (power of two, up to 32).
    int nchunks = 32;
    while (nchunks > 1 &&
           (size_t)BATCH * nchunks * 272 * sizeof(float) > ws_size) {
        nchunks >>= 1;
    }

    float* Gp = (float*)d_ws;                                   // [B][nchunks][256]
    float* Sp = Gp + (size_t)BATCH * nchunks * 256;             // [B][nchunks][16]

    dim3 grid(nchunks, BATCH);
    dice_gram_kernel<<<grid, 256, 0, stream>>>(am, Gp, Sp, nchunks);
    dice_finalize_kernel<<<1, 256, 0, stream>>>(Gp, Sp, out, nchunks);
}